// HODLRLoss_43559558316162
// MI455X (gfx1250) — compile-verified
//
#include <hip/hip_runtime.h>
#include <hip/hip_bf16.h>
#include <math.h>

// Problem constants (match reference file)
#define N_NODES   4194304u        // 1<<22
#define LEVELS    10
#define COARSE    8192
#define NNZ_      8388608u
#define INV_M     (1.0f/512.0f)   // COARSE / N_NODES

typedef __attribute__((ext_vector_type(2))) float v2f;
typedef __attribute__((ext_vector_type(8))) float v8f;

// ---------------------------------------------------------------------------
// Wave32 sum-reduction on the matrix pipe using V_WMMA_F32_16X16X4_F32.
// Step 1: A = (x, 0) per lane, B = ones  ->  D_r = x[r] + x[r+16]   (rowsums)
// Step 2: four accumulating WMMAs feed (D0,D1)..(D6,D7) back as A, summing all
//         16 rowsums -> every element of C holds the full 32-lane total.
// ---------------------------------------------------------------------------
static __device__ __forceinline__ float wave_reduce_sum_wmma(float x) {
  const v2f ones = {1.0f, 1.0f};
  v2f a = {x, 0.0f};
  v8f c0 = {0.f, 0.f, 0.f, 0.f, 0.f, 0.f, 0.f, 0.f};
  v8f d = __builtin_amdgcn_wmma_f32_16x16x4_f32(false, a, false, ones,
                                                (short)0, c0, false, false);
  v8f c = {0.f, 0.f, 0.f, 0.f, 0.f, 0.f, 0.f, 0.f};
#pragma unroll
  for (int q = 0; q < 4; ++q) {
    v2f p = {d[2 * q], d[2 * q + 1]};
    c = __builtin_amdgcn_wmma_f32_16x16x4_f32(false, p, false, ones,
                                              (short)0, c, false, false);
  }
  return c[0];
}

// clamped fractional coarse position (matches F.interpolate align_corners=False)
static __device__ __forceinline__ float posclamp(unsigned i) {
  float pos = ((float)i + 0.5f) * INV_M - 0.5f;
  return fminf(fmaxf(pos, 0.0f), (float)(COARSE - 1));
}

// ---------------------------------------------------------------------------
// Kernel 0: zero y (SpMV accumulator), s (projection table), out (loss)
// ---------------------------------------------------------------------------
__global__ void k_zero(float* __restrict__ y, float* __restrict__ s,
                       float* __restrict__ out) {
  unsigned i = blockIdx.x * 256u + threadIdx.x;
  if (i < N_NODES) y[i] = 0.0f;
  if (i < 4096u) s[i] = 0.0f;
  if (i == 0u) out[0] = 0.0f;
}

// ---------------------------------------------------------------------------
// Kernel 1: fused rank-2 projections  s[lvl][half][r] = sum_i v_full[i]*z[i].
// Per thread (16 contiguous elements) the interp index i0 changes at most
// once, so only coarse cells {i0f, i0f+1, i0f+2} are touched.  The lerp
// factors out: accumulate level-independent weight sums S0,S1,T0,T1, then per
// level the projection is  c0*S0 + c1*(S1+T0) + c2*T1  (3 float2 loads, 6 FMA).
// Wave totals reduced on the matrix pipe.
// ---------------------------------------------------------------------------
__global__ void __launch_bounds__(256) k_hodlr_partials(
    const float* __restrict__ z, const float* __restrict__ vfac,
    float* __restrict__ s) {
  const unsigned base = blockIdx.x * 4096u;
  const unsigned ibase = base + threadIdx.x * 16u;
  const float2* vf2 = reinterpret_cast<const float2*>(vfac);

  const int i0f = (int)posclamp(ibase);
  const int c1i = min(i0f + 1, COARSE - 1);
  const int c2i = min(i0f + 2, COARSE - 1);

  // level-independent weighted sums of z over this thread's 16 elements
  float S0 = 0.f, S1 = 0.f, T0 = 0.f, T1 = 0.f;
  const float4* z4 = reinterpret_cast<const float4*>(z + ibase);
#pragma unroll
  for (int q = 0; q < 4; ++q) {
    float4 t = z4[q];
    float zq[4] = {t.x, t.y, t.z, t.w};
#pragma unroll
    for (int e = 0; e < 4; ++e) {
      const unsigned i = ibase + (unsigned)(q * 4 + e);
      float pos = posclamp(i);
      int i0 = (int)pos;
      float fw = pos - (float)i0;
      float zi = zq[e];
      float g1 = fw * zi;
      float g0 = zi - g1;
      if (i0 == i0f) { S0 += g0; S1 += g1; } else { T0 += g0; T1 += g1; }
    }
  }
  const float m1 = S1 + T0;  // combined coefficient of c1

  float acc[2 * LEVELS];
#pragma unroll
  for (int lvl = 0; lvl < LEVELS; ++lvl) {
    float2 c0 = vf2[lvl * COARSE + i0f];
    float2 c1 = vf2[lvl * COARSE + c1i];
    float2 c2 = vf2[lvl * COARSE + c2i];
    acc[2 * lvl]     = c0.x * S0 + c1.x * m1 + c2.x * T1;
    acc[2 * lvl + 1] = c0.y * S0 + c1.y * m1 + c2.y * T1;
  }

  const unsigned lane = threadIdx.x & 31u;
#pragma unroll
  for (int j = 0; j < 2 * LEVELS; ++j) {
    float tot = wave_reduce_sum_wmma(acc[j]);   // EXEC all-1s here
    if (lane == 0u) {
      const int lvl = j >> 1;
      const int r = j & 1;
      const unsigned h = base >> (21 - lvl);               // half index
      const unsigned idx = 4u * ((1u << lvl) - 1u) + h * 2u + (unsigned)r;
      unsafeAtomicAdd(&s[idx], tot);
    }
  }
}

// ---------------------------------------------------------------------------
// Kernel 2: w[i] = diag*z + sum_lvl lerp(U)[i] . s[lvl][half(i)^1].
// The sibling projection s is block-uniform (scalar loads); fold all 10 levels
// into 3 per-thread scalars K0,K1,K2, then each element is diag*z + lerp(K).
// ---------------------------------------------------------------------------
__global__ void __launch_bounds__(256) k_hodlr_apply(
    const float* __restrict__ diag, const float* __restrict__ z,
    const float* __restrict__ ufac, const float* __restrict__ s,
    float* __restrict__ w) {
  const unsigned base = blockIdx.x * 4096u;
  const unsigned ibase = base + threadIdx.x * 16u;
  const float2* uf2 = reinterpret_cast<const float2*>(ufac);
  const float2* s2 = reinterpret_cast<const float2*>(s);

  const int i0f = (int)posclamp(ibase);
  const int c1i = min(i0f + 1, COARSE - 1);
  const int c2i = min(i0f + 2, COARSE - 1);

  float K0 = 0.f, K1 = 0.f, K2 = 0.f;
#pragma unroll
  for (int lvl = 0; lvl < LEVELS; ++lvl) {
    const unsigned h = base >> (21 - lvl);
    float2 sp = s2[2u * ((1u << lvl) - 1u) + (h ^ 1u)];  // sibling projection
    float2 c0 = uf2[lvl * COARSE + i0f];
    float2 c1 = uf2[lvl * COARSE + c1i];
    float2 c2 = uf2[lvl * COARSE + c2i];
    K0 += c0.x * sp.x + c0.y * sp.y;
    K1 += c1.x * sp.x + c1.y * sp.y;
    K2 += c2.x * sp.x + c2.y * sp.y;
  }

  const float4* z4 = reinterpret_cast<const float4*>(z + ibase);
  const float4* d4 = reinterpret_cast<const float4*>(diag + ibase);
#pragma unroll
  for (int q = 0; q < 4; ++q) {
    float4 zt = z4[q];
    float4 dt = d4[q];
    float zq[4] = {zt.x, zt.y, zt.z, zt.w};
    float dq[4] = {dt.x, dt.y, dt.z, dt.w};
    float o[4];
#pragma unroll
    for (int e = 0; e < 4; ++e) {
      const unsigned i = ibase + (unsigned)(q * 4 + e);
      float pos = posclamp(i);
      int i0 = (int)pos;
      float fw = pos - (float)i0;
      bool sel = (i0 != i0f);
      float A = sel ? K1 : K0;
      float B = sel ? K2 : K1;
      o[e] = dq[e] * zq[e] + A + fw * (B - A);
    }
    *reinterpret_cast<float4*>(w + ibase + 4u * q) =
        make_float4(o[0], o[1], o[2], o[3]);
  }
}

// ---------------------------------------------------------------------------
// Kernel 3: sparse y[row] += A_val * w[col]  (native f32 global atomics; w and
// y are L2-resident at 16 MB each, so gathers/scatters stay on-chip)
// ---------------------------------------------------------------------------
__global__ void k_spmv(const float* __restrict__ vals,
                       const int* __restrict__ rows,
                       const int* __restrict__ cols,
                       const float* __restrict__ w, float* __restrict__ y) {
  unsigned k = blockIdx.x * blockDim.x + threadIdx.x;
  const unsigned stride = gridDim.x * blockDim.x;
  for (; k < NNZ_; k += stride) {
    unsafeAtomicAdd(&y[rows[k]], vals[k] * w[cols[k]]);
  }
}

// ---------------------------------------------------------------------------
// Kernel 4: loss = mean((y - z)^2); wave sums on the matrix pipe.
// ---------------------------------------------------------------------------
__global__ void __launch_bounds__(256) k_loss(const float* __restrict__ y,
                                              const float* __restrict__ z,
                                              float* __restrict__ out) {
  const unsigned ibase = (blockIdx.x * 256u + threadIdx.x) * 8u;
  float acc = 0.0f;
#pragma unroll
  for (int q = 0; q < 2; ++q) {
    float4 yq = *reinterpret_cast<const float4*>(y + ibase + 4u * q);
    float4 zq = *reinterpret_cast<const float4*>(z + ibase + 4u * q);
    float d0 = yq.x - zq.x, d1 = yq.y - zq.y;
    float d2 = yq.z - zq.z, d3 = yq.w - zq.w;
    acc += d0 * d0 + d1 * d1 + d2 * d2 + d3 * d3;
  }
  float tot = wave_reduce_sum_wmma(acc);        // EXEC all-1s, no tails
  if ((threadIdx.x & 31u) == 0u)
    unsafeAtomicAdd(out, tot * (1.0f / (float)N_NODES));
}

// ---------------------------------------------------------------------------
extern "C" void kernel_launch(void* const* d_in, const int* in_sizes, int n_in,
                              void* d_out, int out_size, void* d_ws,
                              size_t ws_size, hipStream_t stream) {
  // setup_inputs order: diag, u_factors, v_factors, A_values, z, A_indices,
  // num_nodes
  const float* diag = (const float*)d_in[0];
  const float* ufac = (const float*)d_in[1];
  const float* vfac = (const float*)d_in[2];
  const float* vals = (const float*)d_in[3];
  const float* z = (const float*)d_in[4];
  const int* aidx = (const int*)d_in[5];
  float* out = (float*)d_out;

  // workspace layout: w (N), y (N), s (4096)
  float* w = (float*)d_ws;
  float* y = w + N_NODES;
  float* s = y + N_NODES;
  const int* rows = aidx;
  const int* cols = aidx + NNZ_;

  k_zero<<<N_NODES / 256u, 256, 0, stream>>>(y, s, out);
  k_hodlr_partials<<<N_NODES / 4096u, 256, 0, stream>>>(z, vfac, s);
  k_hodlr_apply<<<N_NODES / 4096u, 256, 0, stream>>>(diag, z, ufac, s, w);
  k_spmv<<<8192, 256, 0, stream>>>(vals, rows, cols, w, y);
  k_loss<<<N_NODES / (256u * 8u), 256, 0, stream>>>(y, z, out);
}